// MyCurEvaluation_33758442946606
// MI455X (gfx1250) — compile-verified
//
#include <hip/hip_runtime.h>

// ---------------------------------------------------------------------------
// Problem geometry (from reference): [B=64, S=100, K=4, H=32, W=32] f32,
// reduce mean |convdata[:,:,3]-output[:,:,3]| over B*S*H*W = 6,553,600 elems.
// Memory-bound: ~52.4 MB read @ 23.3 TB/s => ~2.3 us. Use b128 loads +
// V_WMMA_F32_16X16X4_F32 as the wave32 cross-lane reducer.
// ---------------------------------------------------------------------------

typedef __attribute__((ext_vector_type(2))) float v2f;
typedef __attribute__((ext_vector_type(8))) float v8f;

#define NBS            6400      // B*S slices
#define SLICE_ELEMS    1024      // H*W
#define BS_STRIDE      4096      // K*H*W elements between (b,s) pairs
#define CH3_OFF        3072      // channel-3 offset within a (b,s) pair
#define SLICES_PER_BLK 8
#define THREADS        256
#define NBLOCKS        (NBS / SLICES_PER_BLK)   // 800
#define TOTAL_ELEMS    6553600.0f

// Reduce a per-lane float across the wave32 using one WMMA:
//   A(16x4): lane m -> A[m][0] (m<16), lane m+16 -> A[m][2]; K=1,3 are zero.
//   B(4x16) = ones, C = 0  =>  D[m][n] = acc_m + acc_{m+16} for all n.
//   Lanes 0-15 hold rows 0-7 in d[0..7]; lanes 16-31 hold rows 8-15.
//   In-lane sum of d[0..7] + shfl_xor(16) = full 32-lane sum in every lane.
// EXEC must be all ones at the call site (no divergence before this point).
__device__ __forceinline__ float wave_reduce_wmma(float acc) {
    v2f a; a.s0 = acc;  a.s1 = 0.0f;
    v2f b; b.s0 = 1.0f; b.s1 = 1.0f;
    v8f c = {};
    v8f d = __builtin_amdgcn_wmma_f32_16x16x4_f32(
        /*neg_a=*/false, a, /*neg_b=*/false, b,
        /*c_mod=*/(short)0, c, /*reuse_a=*/false, /*reuse_b=*/false);
    float s = d.s0 + d.s1 + d.s2 + d.s3 + d.s4 + d.s5 + d.s6 + d.s7;
    s += __shfl_xor(s, 16, 32);
    return s;
}

__global__ __launch_bounds__(THREADS)
void absdiff_partial_kernel(const float* __restrict__ out,
                            const float* __restrict__ conv,
                            float* __restrict__ partials) {
    const int tid = threadIdx.x;
    const int slice0 = blockIdx.x * SLICES_PER_BLK;

    float acc = 0.0f;
#pragma unroll
    for (int s = 0; s < SLICES_PER_BLK; ++s) {
        const long long base =
            (long long)(slice0 + s) * BS_STRIDE + CH3_OFF + (long long)tid * 4;
        // Prefetch next slice's lines while consuming this one.
        if (s + 1 < SLICES_PER_BLK) {
            __builtin_prefetch(out  + base + BS_STRIDE, 0, 3);
            __builtin_prefetch(conv + base + BS_STRIDE, 0, 3);
        }
        const float4 o = *reinterpret_cast<const float4*>(out  + base);
        const float4 c = *reinterpret_cast<const float4*>(conv + base);
        acc += fabsf(c.x - o.x) + fabsf(c.y - o.y)
             + fabsf(c.z - o.z) + fabsf(c.w - o.w);
    }

    const float wsum = wave_reduce_wmma(acc);   // all lanes converged here

    __shared__ float lds[THREADS / 32];
    const int wave = tid >> 5;
    if ((tid & 31) == 0) lds[wave] = wsum;
    __syncthreads();
    if (tid == 0) {
        float bsum = 0.0f;
#pragma unroll
        for (int w = 0; w < THREADS / 32; ++w) bsum += lds[w];
        partials[blockIdx.x] = bsum;
    }
}

__global__ __launch_bounds__(THREADS)
void final_reduce_kernel(const float* __restrict__ partials,
                         float* __restrict__ result) {
    const int tid = threadIdx.x;
    float acc = 0.0f;
    for (int i = tid; i < NBLOCKS; i += THREADS) acc += partials[i];

    const float wsum = wave_reduce_wmma(acc);   // loop re-converges; EXEC all-1s

    __shared__ float lds[THREADS / 32];
    const int wave = tid >> 5;
    if ((tid & 31) == 0) lds[wave] = wsum;
    __syncthreads();
    if (tid == 0) {
        float total = 0.0f;
#pragma unroll
        for (int w = 0; w < THREADS / 32; ++w) total += lds[w];
        result[0] = total * (1.0f / TOTAL_ELEMS);
    }
}

extern "C" void kernel_launch(void* const* d_in, const int* in_sizes, int n_in,
                              void* d_out, int out_size, void* d_ws, size_t ws_size,
                              hipStream_t stream) {
    const float* out_t  = (const float*)d_in[0];   // "output"
    const float* conv_t = (const float*)d_in[1];   // "convdata"
    float* result   = (float*)d_out;               // scalar f32
    float* partials = (float*)d_ws;                // needs NBLOCKS*4 = 3200 B

    absdiff_partial_kernel<<<NBLOCKS, THREADS, 0, stream>>>(out_t, conv_t, partials);
    final_reduce_kernel<<<1, THREADS, 0, stream>>>(partials, result);
}